// MaskedSequenceTransformerOverlap_89275190215502
// MI455X (gfx1250) — compile-verified
//
#include <hip/hip_runtime.h>
#include <math.h>

#define BB   16
#define SS   16
#define VV   64
#define EE   64
#define HH   8
#define NLL  4
#define DFFD 2048
#define LL   (SS * VV)   // 1024
#define DH   (EE / HH)   // 8

typedef __attribute__((ext_vector_type(2))) float v2f;
typedef __attribute__((ext_vector_type(8))) float v8f;

// D = A(16x4 f32) x B(4x16 f32) + C(16x16 f32), full precision matrix op.
static __device__ __forceinline__ v8f wmma4(v2f a, v2f b, v8f c) {
  return __builtin_amdgcn_wmma_f32_16x16x4_f32(
      /*neg_a=*/false, a, /*neg_b=*/false, b,
      /*c_mod=*/(short)0, c, /*reuse_a=*/false, /*reuse_b=*/false);
}

// ---------------------------------------------------------------------------
// Embedding: x[tok,e] = (concat(feat@W_raw+b_raw, overlap_emb, poi_emb) @ W_comb
//                        + b_comb) + pos_tab[s] + type_tab[v];  zm = zero-token mask
// ---------------------------------------------------------------------------
__global__ void embed_kernel(const float* __restrict__ features,
                             const int*   __restrict__ overlap_tag,
                             const int*   __restrict__ poi_id,
                             const float* __restrict__ W_raw,
                             const float* __restrict__ b_raw,
                             const float* __restrict__ pos_tab,
                             const float* __restrict__ type_tab,
                             const float* __restrict__ poi_tab,
                             const float* __restrict__ overlap_tab,
                             const float* __restrict__ W_comb,
                             const float* __restrict__ b_comb,
                             float* __restrict__ x,
                             int*   __restrict__ zm) {
  int tok = blockIdx.x;          // [0, B*L)
  int e   = threadIdx.x;         // [0, 64)
  int l   = tok % LL;
  int s   = l / VV;
  int v   = l % VV;
  __shared__ float cat[80];
  float f0 = features[tok * 3 + 0];
  float f1 = features[tok * 3 + 1];
  float f2 = features[tok * 3 + 2];
  cat[e] = b_raw[e] + f0 * W_raw[e] + f1 * W_raw[EE + e] + f2 * W_raw[2 * EE + e];
  if (e < 8) {
    int ov = overlap_tag[tok]; if (ov < 0) ov = 0;
    cat[64 + e] = (ov == 0) ? 0.0f : overlap_tab[ov * 8 + e];   // padding_idx=0
  } else if (e < 16) {
    int pi = poi_id[tok]; if (pi < 0) pi = 0;
    cat[72 + (e - 8)] = (pi == 0) ? 0.0f : poi_tab[pi * 8 + (e - 8)];
  }
  if (e == 0) zm[tok] = (f0 + f1 + f2 == 0.0f) ? 1 : 0;
  __syncthreads();
  float acc = b_comb[e];
  #pragma unroll
  for (int j = 0; j < 80; ++j) acc += cat[j] * W_comb[j * EE + e];
  x[tok * EE + e] = acc + pos_tab[s * EE + e] + type_tab[v * EE + e];
}

// ---------------------------------------------------------------------------
// Generic WMMA f32 GEMM: C[M,N] = A[M,K] @ W[K,N] + bias (+ residual).
// One wave per 16x16 output tile; 4 waves/block spread along N.
// ---------------------------------------------------------------------------
__global__ void gemm_wmma_kernel(const float* __restrict__ A,
                                 const float* __restrict__ W,
                                 const float* __restrict__ bias,
                                 const float* __restrict__ residual,
                                 float* __restrict__ C,
                                 int N, int K) {
  int lane  = threadIdx.x & 31;
  int wid   = threadIdx.x >> 5;
  int ntile = blockIdx.x * 4 + wid;
  if (ntile * 16 >= N) return;       // wave-uniform guard (EXEC stays full)
  int mtile = blockIdx.y;
  int m     = lane & 15;
  int koff  = (lane >> 4) * 2;       // lanes 0-15 -> K=0,1 ; lanes 16-31 -> K=2,3
  const float* Arow = A + (size_t)(mtile * 16 + m) * K;
  int col = ntile * 16 + m;
  v8f c = {};
  for (int k = 0; k < K; k += 4) {
    v2f a, b;
    a.x = Arow[k + koff];
    a.y = Arow[k + koff + 1];
    b.x = W[(size_t)(k + koff) * N + col];
    b.y = W[(size_t)(k + koff + 1) * N + col];
    c = wmma4(a, b, c);
  }
  float bcol = bias ? bias[col] : 0.0f;
  int rowbase = mtile * 16 + (lane >> 4) * 8;   // C layout: VGPR r -> row r / r+8
  #pragma unroll
  for (int r = 0; r < 8; ++r) {
    int row = rowbase + r;
    float val = c[r] + bcol;
    if (residual) val += residual[(size_t)row * N + col];
    C[(size_t)row * N + col] = val;
  }
}

// ---------------------------------------------------------------------------
// Fused attention, one workgroup per (b,h). K/V staged through LDS in 512-row
// chunks; online softmax. Mask: allowed iff (t_i==t_j || v_i==v_j) and neither
// token is a zero-token.
// ---------------------------------------------------------------------------
#define CHUNK 512
__global__ void attn_kernel(const float* __restrict__ qkv,
                            const int*   __restrict__ zm,
                            float* __restrict__ o) {
  int h = blockIdx.x, b = blockIdx.y;
  int tid = threadIdx.x;   // 256 threads, 4 query rows each
  __shared__ float ks_[CHUNK * DH];
  __shared__ float vs_[CHUNK * DH];
  __shared__ int   zms_[CHUNK];
  const float scale = 0.35355339059327373f;   // 1/sqrt(DH)
  const float* base = qkv + (size_t)b * LL * 192;

  float q[4][DH], m[4], se[4], acc[4][DH];
  int ti_[4], vi_[4], zi_[4];
  #pragma unroll
  for (int r = 0; r < 4; ++r) {
    int i = tid + r * 256;
    ti_[r] = i >> 6; vi_[r] = i & 63;
    zi_[r] = zm[b * LL + i];
    m[r] = -1e30f; se[r] = 0.0f;
    #pragma unroll
    for (int d = 0; d < DH; ++d) {
      q[r][d]   = base[(size_t)i * 192 + h * DH + d];
      acc[r][d] = 0.0f;
    }
  }
  for (int c0 = 0; c0 < LL; c0 += CHUNK) {
    __syncthreads();
    for (int idx = tid; idx < CHUNK * DH; idx += 256) {
      int j = c0 + idx / DH, d = idx % DH;
      ks_[idx] = base[(size_t)j * 192 + 64  + h * DH + d];
      vs_[idx] = base[(size_t)j * 192 + 128 + h * DH + d];
    }
    for (int idx = tid; idx < CHUNK; idx += 256)
      zms_[idx] = zm[b * LL + c0 + idx];
    __syncthreads();
    for (int j = 0; j < CHUNK; ++j) {
      int jj = c0 + j;
      int tj = jj >> 6, vj = jj & 63;
      int zj = zms_[j];
      #pragma unroll
      for (int r = 0; r < 4; ++r) {
        bool ok = ((ti_[r] == tj) || (vi_[r] == vj)) && !zj && !zi_[r];
        if (!ok) continue;
        float s = 0.0f;
        #pragma unroll
        for (int d = 0; d < DH; ++d) s += q[r][d] * ks_[j * DH + d];
        s *= scale;
        if (s > m[r]) {
          float corr = __expf(m[r] - s);
          se[r] *= corr;
          #pragma unroll
          for (int d = 0; d < DH; ++d) acc[r][d] *= corr;
          m[r] = s;
        }
        float p = __expf(s - m[r]);
        se[r] += p;
        #pragma unroll
        for (int d = 0; d < DH; ++d) acc[r][d] += p * vs_[j * DH + d];
      }
    }
  }
  #pragma unroll
  for (int r = 0; r < 4; ++r) {
    int i = tid + r * 256;
    float inv = (se[r] > 0.0f) ? 1.0f / se[r] : 0.0f;
    #pragma unroll
    for (int d = 0; d < DH; ++d)
      o[((size_t)b * LL + i) * EE + h * DH + d] = acc[r][d] * inv;
  }
}

// ---------------------------------------------------------------------------
// LayerNorm over E=64, one 64-thread block per token.
// ---------------------------------------------------------------------------
__global__ void ln_kernel(const float* __restrict__ in,
                          const float* __restrict__ gain,
                          const float* __restrict__ bias,
                          float* __restrict__ out) {
  int tok = blockIdx.x, e = threadIdx.x;
  __shared__ float red[64];
  float v = in[(size_t)tok * EE + e];
  red[e] = v; __syncthreads();
  for (int s = 32; s > 0; s >>= 1) { if (e < s) red[e] += red[e + s]; __syncthreads(); }
  float mean = red[0] / 64.0f;
  __syncthreads();
  float d = v - mean;
  red[e] = d * d; __syncthreads();
  for (int s = 32; s > 0; s >>= 1) { if (e < s) red[e] += red[e + s]; __syncthreads(); }
  float var = red[0] / 64.0f;
  out[(size_t)tok * EE + e] = d * rsqrtf(var + 1e-5f) * gain[e] + bias[e];
}

// ---------------------------------------------------------------------------
// Fused FFN: out = x + relu(x @ W1 + b1) @ W2 + b2, DFF never hits HBM.
// One wave per 16-row tile; DFF consumed in 16-wide slices. The hidden tile is
// produced in WMMA C-layout, bounced through LDS, and re-read in A-layout for
// the second GEMM (wave-local, ordered by s_wait_dscnt).
// ---------------------------------------------------------------------------
__global__ void ffn_kernel(const float* __restrict__ x,
                           const float* __restrict__ W1,
                           const float* __restrict__ b1,
                           const float* __restrict__ W2,
                           const float* __restrict__ b2,
                           float* __restrict__ out) {
  int lane  = threadIdx.x & 31;
  int wid   = threadIdx.x >> 5;
  int mtile = blockIdx.x * 4 + wid;
  __shared__ float hbuf[4][16][17];     // +1 pad to dodge LDS bank conflicts
  int m    = lane & 15;
  int koff = (lane >> 4) * 2;
  const float* xrow = x + (size_t)(mtile * 16 + m) * EE;

  v2f ax[16];                            // x-tile A fragments, reused 128 times
  #pragma unroll
  for (int ks = 0; ks < 16; ++ks) {
    ax[ks].x = xrow[ks * 4 + koff];
    ax[ks].y = xrow[ks * 4 + koff + 1];
  }
  v8f acc[4] = {};                       // 16x64 output accumulators

  for (int nf = 0; nf < DFFD / 16; ++nf) {
    // h = x_tile @ W1[:, nf*16 .. nf*16+16)
    v8f hc = {};
    int hcol = nf * 16 + m;
    #pragma unroll
    for (int ks = 0; ks < 16; ++ks) {
      v2f bf;
      bf.x = W1[(size_t)(ks * 4 + koff) * DFFD + hcol];
      bf.y = W1[(size_t)(ks * 4 + koff + 1) * DFFD + hcol];
      hc = wmma4(ax[ks], bf, hc);
    }
    // bias + relu, C-layout -> LDS (row-major 16x16 tile)
    float bb = b1[hcol];
    #pragma unroll
    for (int r = 0; r < 8; ++r) {
      float hv = hc[r] + bb;
      hv = hv > 0.0f ? hv : 0.0f;
      hbuf[wid][r + (lane >> 4) * 8][m] = hv;
    }
    asm volatile("s_wait_dscnt 0" ::: "memory");
    // re-read as A fragments (K = this 16-slice of DFF)
    v2f ah[4];
    #pragma unroll
    for (int k2 = 0; k2 < 4; ++k2) {
      ah[k2].x = hbuf[wid][m][k2 * 4 + koff];
      ah[k2].y = hbuf[wid][m][k2 * 4 + koff + 1];
    }
    asm volatile("s_wait_dscnt 0" ::: "memory");
    // acc += h @ W2[nf*16 .. nf*16+16, :]
    #pragma unroll
    for (int nt = 0; nt < 4; ++nt) {
      #pragma unroll
      for (int k2 = 0; k2 < 4; ++k2) {
        v2f bf;
        int krow = nf * 16 + k2 * 4 + koff;
        bf.x = W2[(size_t)krow * EE + nt * 16 + m];
        bf.y = W2[(size_t)(krow + 1) * EE + nt * 16 + m];
        acc[nt] = wmma4(ah[k2], bf, acc[nt]);
      }
    }
  }
  // epilogue: + b2 + residual x
  #pragma unroll
  for (int nt = 0; nt < 4; ++nt) {
    #pragma unroll
    for (int r = 0; r < 8; ++r) {
      int row = mtile * 16 + r + (lane >> 4) * 8;
      int col = nt * 16 + m;
      out[(size_t)row * EE + col] = acc[nt][r] + b2[col] + x[(size_t)row * EE + col];
    }
  }
}

// ---------------------------------------------------------------------------
// Head: out[b,v,:] = x[b, last step, v, :] @ W_head + b_head
// ---------------------------------------------------------------------------
__global__ void head_kernel(const float* __restrict__ x,
                            const float* __restrict__ W_head,
                            const float* __restrict__ b_head,
                            float* __restrict__ out) {
  int b = blockIdx.x, v = threadIdx.x;
  const float* row = x + ((size_t)b * LL + (SS - 1) * VV + v) * EE;
  float o0 = b_head[0], o1 = b_head[1], o2 = b_head[2];
  #pragma unroll
  for (int e = 0; e < EE; ++e) {
    float xv = row[e];
    o0 += xv * W_head[e * 3 + 0];
    o1 += xv * W_head[e * 3 + 1];
    o2 += xv * W_head[e * 3 + 2];
  }
  size_t base = ((size_t)b * VV + v) * 3;
  out[base + 0] = o0; out[base + 1] = o1; out[base + 2] = o2;
}

// ---------------------------------------------------------------------------
extern "C" void kernel_launch(void* const* d_in, const int* in_sizes, int n_in,
                              void* d_out, int out_size, void* d_ws, size_t ws_size,
                              hipStream_t stream) {
  (void)in_sizes; (void)n_in; (void)out_size; (void)ws_size;
  const float* features    = (const float*)d_in[0];
  const int*   overlap_tag = (const int*)  d_in[1];
  const int*   poi_id      = (const int*)  d_in[2];
  const float* W_raw       = (const float*)d_in[3];
  const float* b_raw       = (const float*)d_in[4];
  const float* pos_tab     = (const float*)d_in[5];
  const float* type_tab    = (const float*)d_in[6];
  const float* poi_tab     = (const float*)d_in[7];
  const float* overlap_tab = (const float*)d_in[8];
  const float* W_comb      = (const float*)d_in[9];
  const float* b_comb      = (const float*)d_in[10];
  const float* Wqkv        = (const float*)d_in[11];
  const float* bqkv        = (const float*)d_in[12];
  const float* Wo          = (const float*)d_in[13];
  const float* bo          = (const float*)d_in[14];
  const float* ln1_s       = (const float*)d_in[15];
  const float* ln1_b       = (const float*)d_in[16];
  const float* W1          = (const float*)d_in[17];
  const float* b1          = (const float*)d_in[18];
  const float* W2          = (const float*)d_in[19];
  const float* b2          = (const float*)d_in[20];
  const float* ln2_s       = (const float*)d_in[21];
  const float* ln2_b       = (const float*)d_in[22];
  const float* W_head      = (const float*)d_in[23];
  const float* b_head      = (const float*)d_in[24];

  const size_t NTOK = (size_t)BB * LL;    // 16384 tokens
  float* ws  = (float*)d_ws;
  float* x   = ws;                        // [NTOK, 64]
  float* y   = x   + NTOK * EE;           // [NTOK, 64]
  float* qkv = y   + NTOK * EE;           // [NTOK, 192]
  float* o   = qkv + NTOK * 3 * EE;       // [NTOK, 64]
  int*   zm  = (int*)(o + NTOK * EE);     // [NTOK]

  embed_kernel<<<(int)NTOK, 64, 0, stream>>>(
      features, overlap_tag, poi_id, W_raw, b_raw, pos_tab, type_tab,
      poi_tab, overlap_tab, W_comb, b_comb, x, zm);

  const int MT = (int)(NTOK / 16);        // 1024 row tiles
  for (int l = 0; l < NLL; ++l) {
    // qkv = x @ Wqkv[l] + bqkv[l]           (M=16384, N=192, K=64)
    gemm_wmma_kernel<<<dim3(3, MT), 128, 0, stream>>>(
        x, Wqkv + (size_t)l * EE * 3 * EE, bqkv + (size_t)l * 3 * EE,
        nullptr, qkv, 3 * EE, EE);
    // fused masked attention
    attn_kernel<<<dim3(HH, BB), 256, 0, stream>>>(qkv, zm, o);
    // y = x + o @ Wo[l] + bo[l]             (M=16384, N=64, K=64)
    gemm_wmma_kernel<<<dim3(1, MT), 128, 0, stream>>>(
        o, Wo + (size_t)l * EE * EE, bo + (size_t)l * EE, x, y, EE, EE);
    // x = LN1(y)
    ln_kernel<<<(int)NTOK, 64, 0, stream>>>(y, ln1_s + (size_t)l * EE,
                                            ln1_b + (size_t)l * EE, x);
    // y = x + relu(x@W1+b1)@W2 + b2 (fused, DFF stays on-chip)
    ffn_kernel<<<MT / 4, 128, 0, stream>>>(
        x, W1 + (size_t)l * EE * DFFD, b1 + (size_t)l * DFFD,
        W2 + (size_t)l * DFFD * EE, b2 + (size_t)l * EE, y);
    // x = LN2(y)
    ln_kernel<<<(int)NTOK, 64, 0, stream>>>(y, ln2_s + (size_t)l * EE,
                                            ln2_b + (size_t)l * EE, x);
  }
  head_kernel<<<BB, VV, 0, stream>>>(x, W_head, b_head, (float*)d_out);
}